// GraphAttentionLayer_83184926589020
// MI455X (gfx1250) — compile-verified
//
#include <hip/hip_runtime.h>
#include <hip/hip_bf16.h>

#define N_NODES 50000
#define N_EDGES 1600000
#define D_FEAT  128
#define NEG_SLOPE 0.2f

typedef __attribute__((ext_vector_type(2))) float v2f;
typedef __attribute__((ext_vector_type(8))) float v8f;

__device__ __forceinline__ float leaky(float x) {
    return x >= 0.0f ? x : NEG_SLOPE * x;
}

// order-preserving float -> uint transform (monotonic), for atomic segment-max
__device__ __forceinline__ unsigned f2ord(float f) {
    unsigned u = __float_as_uint(f);
    return (u & 0x80000000u) ? ~u : (u | 0x80000000u);
}
__device__ __forceinline__ float ord2f(unsigned o) {
    unsigned u = (o & 0x80000000u) ? (o & 0x7FFFFFFFu) : ~o;
    return __uint_as_float(u);
}

// ---------------------------------------------------------------------------
// zero helpers (d_out / reduction buffers are poisoned by the harness)
// ---------------------------------------------------------------------------
__global__ void zero_f32(float* __restrict__ p, long n) {
    long i = (long)blockIdx.x * blockDim.x + threadIdx.x;
    long stride = (long)gridDim.x * blockDim.x;
    for (; i < n; i += stride) p[i] = 0.0f;
}

__global__ void zero_md(unsigned* __restrict__ m_ord, float* __restrict__ denom, int n) {
    int i = blockIdx.x * blockDim.x + threadIdx.x;
    if (i < n) { m_ord[i] = 0u; denom[i] = 0.0f; }  // ord 0 == -inf-like sentinel
}

// ---------------------------------------------------------------------------
// GEMM: out[r, cb + 0:128] = leaky( Z[r, 0:128] @ W + b )
// Z row stride ldz, out row stride ldo, out column base cb.
// Block = 256 threads (8 waves). Block -> 16 rows; wave w -> 16-col tile.
// Uses V_WMMA_F32_16X16X4_F32, K accumulated over 32 steps of 4.
// ---------------------------------------------------------------------------
#define LDS_STRIDE 132  // 128 + 4 pad: conflict-free A-fragment reads

__global__ __launch_bounds__(256)
void gemm_bias_leaky(const float* __restrict__ Z, int ldz,
                     const float* __restrict__ W,   // [128 x 128] row-major (k, n)
                     const float* __restrict__ b,
                     float* __restrict__ out, int ldo, int cb) {
    __shared__ float As[16 * LDS_STRIDE];

    const int tid = threadIdx.x;
    const int rowbase = blockIdx.x * 16;

    // Stage A tile: 16 rows x 128 cols, 8 floats (2x float4) per thread.
    {
        const int r = tid >> 4;            // 0..15
        const int c = (tid & 15) * 8;      // 0,8,...,120
        const float4* src = (const float4*)(Z + (long)(rowbase + r) * ldz + c);
        float4 v0 = src[0];
        float4 v1 = src[1];
        float* dst = &As[r * LDS_STRIDE + c];
        dst[0] = v0.x; dst[1] = v0.y; dst[2] = v0.z; dst[3] = v0.w;
        dst[4] = v1.x; dst[5] = v1.y; dst[6] = v1.z; dst[7] = v1.w;
    }
    __syncthreads();

    const int wave = tid >> 5;             // 0..7 -> 16-col tile
    const int lane = tid & 31;
    const int mn   = lane & 15;            // A: row within tile; B/D: col within tile
    const int kg   = (lane >> 4) * 2;      // K sub-offset 0 or 2
    const int ncol = wave * 16 + mn;       // absolute col in [0,128)

    v8f acc = {};

#if __has_builtin(__builtin_amdgcn_wmma_f32_16x16x4_f32)
    #pragma unroll
    for (int k = 0; k < 128; k += 4) {
        v2f a, bb;
        a.x = As[mn * LDS_STRIDE + k + kg + 0];
        a.y = As[mn * LDS_STRIDE + k + kg + 1];
        bb.x = W[(k + kg + 0) * 128 + ncol];
        bb.y = W[(k + kg + 1) * 128 + ncol];
        acc = __builtin_amdgcn_wmma_f32_16x16x4_f32(
            false, a, false, bb, (short)0, acc, false, false);
    }
#else
    // scalar fallback (should not trigger on gfx1250)
    #pragma unroll 4
    for (int k = 0; k < 128; ++k) {
        float a = As[mn * LDS_STRIDE + k];
        #pragma unroll
        for (int v = 0; v < 8; ++v) acc[v] += a * W[k * 128 + ncol];
    }
#endif

    // D layout: VGPR v holds row (v + 8*(lane/16)), col = lane%16
    const float bias = b[ncol];
    const int rhalf = (lane >> 4) * 8;
    #pragma unroll
    for (int v = 0; v < 8; ++v) {
        float val = leaky(acc[v] + bias);
        out[(long)(rowbase + rhalf + v) * ldo + cb + ncol] = val;
    }
}

// ---------------------------------------------------------------------------
// Per-node attention scores: s1[n] = z[n]·a1, s2[n] = z[n]·a2
// One wave per node; lane handles 4 contiguous floats; shfl_xor reduce.
// ---------------------------------------------------------------------------
__global__ __launch_bounds__(256)
void score_kernel(const float* __restrict__ Z, int ldz,
                  const float* __restrict__ a1, const float* __restrict__ a2,
                  float* __restrict__ s1, float* __restrict__ s2, int n_nodes) {
    const int wave = threadIdx.x >> 5;
    const int lane = threadIdx.x & 31;
    const int node = blockIdx.x * 8 + wave;
    if (node >= n_nodes) return;

    const float4 zv = *(const float4*)(Z + (long)node * ldz + lane * 4);
    const float4 av = *(const float4*)(a1 + lane * 4);
    const float4 bv = *(const float4*)(a2 + lane * 4);
    float d1 = zv.x * av.x + zv.y * av.y + zv.z * av.z + zv.w * av.w;
    float d2 = zv.x * bv.x + zv.y * bv.y + zv.z * bv.z + zv.w * bv.w;
    #pragma unroll
    for (int off = 16; off > 0; off >>= 1) {
        d1 += __shfl_xor(d1, off, 32);
        d2 += __shfl_xor(d2, off, 32);
    }
    if (lane == 0) { s1[node] = d1; s2[node] = d2; }
}

// ---------------------------------------------------------------------------
// Edge pass 1: segment max of leaky(s1[src]+s2[dst]) over dst
// ---------------------------------------------------------------------------
__global__ __launch_bounds__(256)
void edge_max_kernel(const int* __restrict__ src, const int* __restrict__ dst,
                     const float* __restrict__ s1, const float* __restrict__ s2,
                     unsigned* __restrict__ m_ord, int n_edges) {
    int i = blockIdx.x * blockDim.x + threadIdx.x;
    if (i >= n_edges) return;
    float e = leaky(s1[src[i]] + s2[dst[i]]);
    atomicMax(&m_ord[dst[i]], f2ord(e));
}

// ---------------------------------------------------------------------------
// Edge pass 2: e_exp = exp(e - max[dst]); denom[dst] += e_exp
// ---------------------------------------------------------------------------
__global__ __launch_bounds__(256)
void edge_exp_kernel(const int* __restrict__ src, const int* __restrict__ dst,
                     const float* __restrict__ s1, const float* __restrict__ s2,
                     const unsigned* __restrict__ m_ord,
                     float* __restrict__ e_exp, float* __restrict__ denom,
                     int n_edges) {
    int i = blockIdx.x * blockDim.x + threadIdx.x;
    if (i >= n_edges) return;
    int d = dst[i];
    float e = leaky(s1[src[i]] + s2[d]);
    float ee = __expf(e - ord2f(m_ord[d]));
    e_exp[i] = ee;
    __hip_atomic_fetch_add(&denom[d], ee, __ATOMIC_RELAXED, __HIP_MEMORY_SCOPE_AGENT);
}

// ---------------------------------------------------------------------------
// Edge pass 3: h[dst, :] += (e_exp/denom[dst]) * z[src, :]
// One wave per edge (grid-stride): lane covers 4 floats -> 128-wide row.
// ---------------------------------------------------------------------------
__global__ __launch_bounds__(256)
void aggregate_kernel(const int* __restrict__ src, const int* __restrict__ dst,
                      const float* __restrict__ e_exp, const float* __restrict__ denom,
                      const float* __restrict__ Z, int ldz,
                      float* __restrict__ H, int ldh, int cb, int n_edges) {
    const int lane = threadIdx.x & 31;
    const int wave_global = blockIdx.x * 8 + (threadIdx.x >> 5);
    const int nwaves = gridDim.x * 8;

    for (int e = wave_global; e < n_edges; e += nwaves) {
        const int s = src[e];
        const int d = dst[e];
        const float alpha = e_exp[e] / denom[d];
        const float4 zv = *(const float4*)(Z + (long)s * ldz + lane * 4);
        float* hp = H + (long)d * ldh + cb + lane * 4;
        __hip_atomic_fetch_add(hp + 0, alpha * zv.x, __ATOMIC_RELAXED, __HIP_MEMORY_SCOPE_AGENT);
        __hip_atomic_fetch_add(hp + 1, alpha * zv.y, __ATOMIC_RELAXED, __HIP_MEMORY_SCOPE_AGENT);
        __hip_atomic_fetch_add(hp + 2, alpha * zv.z, __ATOMIC_RELAXED, __HIP_MEMORY_SCOPE_AGENT);
        __hip_atomic_fetch_add(hp + 3, alpha * zv.w, __ATOMIC_RELAXED, __HIP_MEMORY_SCOPE_AGENT);
    }
}

// ---------------------------------------------------------------------------
// Launch
// ---------------------------------------------------------------------------
extern "C" void kernel_launch(void* const* d_in, const int* in_sizes, int n_in,
                              void* d_out, int out_size, void* d_ws, size_t ws_size,
                              hipStream_t stream) {
    const float* X   = (const float*)d_in[0];
    const int*   src = (const int*)  d_in[1];
    const int*   dst = (const int*)  d_in[2];
    const float* W1  = (const float*)d_in[3];
    const float* b1  = (const float*)d_in[4];
    const float* W2  = (const float*)d_in[5];
    const float* b2  = (const float*)d_in[6];
    const float* a1  = (const float*)d_in[7];
    const float* a2  = (const float*)d_in[8];
    float* out = (float*)d_out;   // [N_NODES, 256]

    // carve workspace
    size_t off = 0;
    auto carve = [&](size_t bytes) -> char* {
        char* p = (char*)d_ws + off;
        off += (bytes + 255) & ~(size_t)255;
        return p;
    };
    float*    h1    = (float*)   carve((size_t)N_NODES * D_FEAT * 4);
    float*    h2    = (float*)   carve((size_t)N_NODES * D_FEAT * 4);
    float*    s1    = (float*)   carve((size_t)N_NODES * 4);
    float*    s2    = (float*)   carve((size_t)N_NODES * 4);
    unsigned* m_ord = (unsigned*)carve((size_t)N_NODES * 4);
    float*    denom = (float*)   carve((size_t)N_NODES * 4);
    float*    e_exp = (float*)   carve((size_t)N_EDGES * 4);

    const int gemm_blocks  = N_NODES / 16;        // 3125
    const int score_blocks = (N_NODES + 7) / 8;   // 6250
    const int edge_blocks  = (N_EDGES + 255) / 256;
    const int agg_blocks   = 8192;
    const int md_blocks    = (N_NODES + 255) / 256;

    // zero the full output (aggregation accumulates into it)
    zero_f32<<<2048, 256, 0, stream>>>(out, (long)N_NODES * 256);

    // ---- round 1 ----
    gemm_bias_leaky<<<gemm_blocks, 256, 0, stream>>>(X, D_FEAT, W1, b1, h1, D_FEAT, 0);
    zero_md<<<md_blocks, 256, 0, stream>>>(m_ord, denom, N_NODES);
    score_kernel<<<score_blocks, 256, 0, stream>>>(h1, D_FEAT, a1, a2, s1, s2, N_NODES);
    edge_max_kernel<<<edge_blocks, 256, 0, stream>>>(src, dst, s1, s2, m_ord, N_EDGES);
    edge_exp_kernel<<<edge_blocks, 256, 0, stream>>>(src, dst, s1, s2, m_ord, e_exp, denom, N_EDGES);
    aggregate_kernel<<<agg_blocks, 256, 0, stream>>>(src, dst, e_exp, denom,
                                                     h1, D_FEAT, out, 256, 0, N_EDGES);

    // ---- round 2 (input = h1' living in out[:, 0:128], row stride 256) ----
    gemm_bias_leaky<<<gemm_blocks, 256, 0, stream>>>(out, 256, W2, b2, h2, D_FEAT, 0);
    zero_md<<<md_blocks, 256, 0, stream>>>(m_ord, denom, N_NODES);
    score_kernel<<<score_blocks, 256, 0, stream>>>(h2, D_FEAT, a1, a2, s1, s2, N_NODES);
    edge_max_kernel<<<edge_blocks, 256, 0, stream>>>(src, dst, s1, s2, m_ord, N_EDGES);
    edge_exp_kernel<<<edge_blocks, 256, 0, stream>>>(src, dst, s1, s2, m_ord, e_exp, denom, N_EDGES);
    aggregate_kernel<<<agg_blocks, 256, 0, stream>>>(src, dst, e_exp, denom,
                                                     h2, D_FEAT, out, 256, 128, N_EDGES);
}